// GCN_87926570483772
// MI455X (gfx1250) — compile-verified
//
#include <hip/hip_runtime.h>
#include <hip/hip_bf16.h>

typedef __attribute__((ext_vector_type(2))) float v2f;
typedef __attribute__((ext_vector_type(8))) float v8f;

#define FEAT 128

// ---------------------------------------------------------------------------
// Degree / normalization
// ---------------------------------------------------------------------------
__global__ void deg_init_kernel(float* __restrict__ deg, int n) {
    int i = blockIdx.x * blockDim.x + threadIdx.x;
    if (i < n) deg[i] = 1.0f;  // self-loop contribution
}

__global__ void deg_edges_kernel(const int* __restrict__ dst, float* __restrict__ deg, int ne) {
    int e = blockIdx.x * blockDim.x + threadIdx.x;
    if (e < ne) unsafeAtomicAdd(&deg[dst[e]], 1.0f);
}

__global__ void deg_to_dinv_kernel(float* __restrict__ deg, int n) {
    int i = blockIdx.x * blockDim.x + threadIdx.x;
    if (i < n) deg[i] = rsqrtf(deg[i]);  // deg >= 1 always (self loops)
}

// ---------------------------------------------------------------------------
// Dense GEMM  H[n,128] = X[n,128] @ W[128,128]  via V_WMMA_F32_16X16X4_F32
// Block: 256 threads = 8 waves covering a 32-row panel. Wave w owns col tile
// n0 = 16*w and BOTH row tiles (two accumulators) so every B fragment is
// reused twice. K marched 4 at a time -> 64 chained WMMAs per wave.
// ---------------------------------------------------------------------------
__global__ __launch_bounds__(256) void gemm128_wmma_kernel(
    const float* __restrict__ X, const float* __restrict__ W,
    float* __restrict__ H, int nrows) {
    __shared__ float As[32][FEAT + 4];  // 32-row A panel, padded

    const int m0 = blockIdx.x * 32;

    // Cooperative load of the 32x128 A panel (zero-pad past nrows)
    for (int i = threadIdx.x; i < 32 * FEAT; i += 256) {
        int r = i >> 7, cc = i & (FEAT - 1);
        int row = m0 + r;
        As[r][cc] = (row < nrows) ? X[(long long)row * FEAT + cc] : 0.0f;
    }
    __syncthreads();

    const int wave  = threadIdx.x >> 5;   // 0..7 -> col tile
    const int lane  = threadIdx.x & 31;
    const int n0    = wave * 16;
    const int mrow  = lane & 15;          // A row within tile
    const int khalf = (lane >> 4) * 2;    // lanes 16-31 hold K+2,K+3
    const int ncol  = n0 + (lane & 15);   // B/C column

    v8f c0 = {};
    v8f c1 = {};
#pragma unroll
    for (int k0 = 0; k0 < FEAT; k0 += 4) {
        v2f a0, a1, b;
        // A 16x4 fragments (ISA 7.12.2: V0 = K, V1 = K+1; upper lanes K+2/K+3)
        a0.x = As[mrow][k0 + khalf];
        a0.y = As[mrow][k0 + khalf + 1];
        a1.x = As[16 + mrow][k0 + khalf];
        a1.y = As[16 + mrow][k0 + khalf + 1];
        // B 4x16 fragment, row-striped across lanes (W is row-major [K,N])
        b.x = W[(k0 + khalf) * FEAT + ncol];
        b.y = W[(k0 + khalf + 1) * FEAT + ncol];
        c0 = __builtin_amdgcn_wmma_f32_16x16x4_f32(
            false, a0, false, b, (short)0, c0, false, false);
        c1 = __builtin_amdgcn_wmma_f32_16x16x4_f32(
            false, a1, false, b, (short)0, c1, false, false);
    }

    // C/D layout: VGPR v holds row (v + 8*(lane>=16)), col = lane&15
    const int mbase = (lane >> 4) * 8;
    float* __restrict__ Hp = H + (long long)(m0 + mbase) * FEAT + ncol;
    if (m0 + 32 <= nrows) {   // uniform fast path: no per-row guards
#pragma unroll
        for (int v = 0; v < 8; ++v)
            Hp[v * FEAT] = c0[v];
#pragma unroll
        for (int v = 0; v < 8; ++v)
            Hp[(16 + v) * FEAT] = c1[v];
    } else {
#pragma unroll
        for (int v = 0; v < 8; ++v) {
            if (m0 + mbase + v < nrows) Hp[v * FEAT] = c0[v];
        }
#pragma unroll
        for (int v = 0; v < 8; ++v) {
            if (m0 + 16 + mbase + v < nrows) Hp[(16 + v) * FEAT] = c1[v];
        }
    }
}

// ---------------------------------------------------------------------------
// Aggregation: out = dinv^2 * h  (self loop, full overwrite -> also the init)
// ---------------------------------------------------------------------------
__global__ void self_init128_kernel(const float* __restrict__ h,
                                    const float* __restrict__ dinv,
                                    float* __restrict__ out, int n) {
    long long i = (long long)blockIdx.x * blockDim.x + threadIdx.x;
    if (i >= (long long)n * FEAT) return;
    int node = (int)(i >> 7);
    float d = dinv[node];
    out[i] = h[i] * d * d;
}

// One block per edge; 128 lanes = 128 features. Atomics land in L2 (both
// 51MB feature buffers fit the 192MB L2).
__global__ __launch_bounds__(128) void edge_agg128_kernel(
    const float* __restrict__ h, const int* __restrict__ src,
    const int* __restrict__ dst, const float* __restrict__ dinv,
    float* __restrict__ out, int ne) {
    int e = blockIdx.x;
    if (e >= ne) return;
    int s = src[e], d = dst[e];
    float w = dinv[s] * dinv[d];
    int f = threadIdx.x;
    unsafeAtomicAdd(&out[(long long)d * FEAT + f],
                    h[(long long)s * FEAT + f] * w);
}

// Fused bias + BN(eval) + ReLU
__global__ void bn_relu_kernel(const float* __restrict__ agg,
                               const float* __restrict__ bias,
                               const float* __restrict__ gamma,
                               const float* __restrict__ beta,
                               const float* __restrict__ mean,
                               const float* __restrict__ var,
                               float* __restrict__ y, int n) {
    long long i = (long long)blockIdx.x * blockDim.x + threadIdx.x;
    if (i >= (long long)n * FEAT) return;
    int f = (int)(i & (FEAT - 1));
    float x = agg[i] + bias[f];
    float v = (x - mean[f]) * rsqrtf(var[f] + 1e-5f) * gamma[f] + beta[f];
    y[i] = fmaxf(v, 0.0f);
}

// ---------------------------------------------------------------------------
// Layer 3: H3[n,2] = Y[n,128] @ W3[128,2]; one wave per node
// ---------------------------------------------------------------------------
__global__ __launch_bounds__(256) void gemm_w3_kernel(
    const float* __restrict__ Y, const float* __restrict__ W3,
    float* __restrict__ H3, int nrows) {
    int node = blockIdx.x * 8 + (threadIdx.x >> 5);
    int lane = threadIdx.x & 31;
    if (node >= nrows) return;
    float s0 = 0.0f, s1 = 0.0f;
#pragma unroll
    for (int f = lane; f < FEAT; f += 32) {
        float x = Y[(long long)node * FEAT + f];
        s0 += x * W3[f * 2 + 0];
        s1 += x * W3[f * 2 + 1];
    }
#pragma unroll
    for (int off = 16; off > 0; off >>= 1) {
        s0 += __shfl_down(s0, off, 32);
        s1 += __shfl_down(s1, off, 32);
    }
    if (lane == 0) {
        H3[node * 2 + 0] = s0;
        H3[node * 2 + 1] = s1;
    }
}

__global__ void self_init2_kernel(const float* __restrict__ h3,
                                  const float* __restrict__ dinv,
                                  float* __restrict__ out, int n) {
    int i = blockIdx.x * blockDim.x + threadIdx.x;
    if (i >= n) return;
    float d = dinv[i];
    float d2 = d * d;
    out[2 * i + 0] = h3[2 * i + 0] * d2;
    out[2 * i + 1] = h3[2 * i + 1] * d2;
}

__global__ void edge_agg2_kernel(const float* __restrict__ h3,
                                 const int* __restrict__ src,
                                 const int* __restrict__ dst,
                                 const float* __restrict__ dinv,
                                 float* __restrict__ out, int ne) {
    int e = blockIdx.x * blockDim.x + threadIdx.x;
    if (e >= ne) return;
    int s = src[e], d = dst[e];
    float w = dinv[s] * dinv[d];
    unsafeAtomicAdd(&out[2 * d + 0], h3[2 * s + 0] * w);
    unsafeAtomicAdd(&out[2 * d + 1], h3[2 * s + 1] * w);
}

__global__ void logsoftmax2_kernel(float* __restrict__ out,
                                   const float* __restrict__ b3, int n) {
    int i = blockIdx.x * blockDim.x + threadIdx.x;
    if (i >= n) return;
    float z0 = out[2 * i + 0] + b3[0];
    float z1 = out[2 * i + 1] + b3[1];
    float mx = fmaxf(z0, z1);
    float lse = mx + logf(expf(z0 - mx) + expf(z1 - mx));
    out[2 * i + 0] = z0 - lse;
    out[2 * i + 1] = z1 - lse;
}

// ---------------------------------------------------------------------------
extern "C" void kernel_launch(void* const* d_in, const int* in_sizes, int n_in,
                              void* d_out, int out_size, void* d_ws, size_t ws_size,
                              hipStream_t stream) {
    const float* x   = (const float*)d_in[0];
    const int*   ei  = (const int*)d_in[1];
    const float* W1  = (const float*)d_in[2];
    const float* b1  = (const float*)d_in[3];
    const float* g1  = (const float*)d_in[4];
    const float* be1 = (const float*)d_in[5];
    const float* m1  = (const float*)d_in[6];
    const float* v1  = (const float*)d_in[7];
    const float* W2  = (const float*)d_in[8];
    const float* b2  = (const float*)d_in[9];
    const float* g2  = (const float*)d_in[10];
    const float* be2 = (const float*)d_in[11];
    const float* m2  = (const float*)d_in[12];
    const float* v2  = (const float*)d_in[13];
    const float* W3  = (const float*)d_in[14];
    const float* b3  = (const float*)d_in[15];

    const int N = in_sizes[0] / FEAT;   // 100000
    const int E = in_sizes[1] / 2;      // 400000
    const int* src = ei;
    const int* dst = ei + E;

    // Workspace layout (floats): dinv[N] | bufA[N*128] | bufB[N*128] | h3[N*2]
    float* dinv = (float*)d_ws;
    float* bufA = dinv + N;
    float* bufB = bufA + (long long)N * FEAT;
    float* h3   = bufB + (long long)N * FEAT;
    float* outp = (float*)d_out;

    const int TB = 256;
    const int gN      = (N + TB - 1) / TB;
    const int gE      = (E + TB - 1) / TB;
    const long long NF = (long long)N * FEAT;
    const int gNF     = (int)((NF + TB - 1) / TB);
    const int gTiles  = (N + 31) / 32;   // 32-row panels
    const int gWave8  = (N + 7) / 8;

    // Normalization
    deg_init_kernel<<<gN, TB, 0, stream>>>(dinv, N);
    deg_edges_kernel<<<gE, TB, 0, stream>>>(dst, dinv, E);
    deg_to_dinv_kernel<<<gN, TB, 0, stream>>>(dinv, N);

    // Layer 1
    gemm128_wmma_kernel<<<gTiles, 256, 0, stream>>>(x, W1, bufA, N);
    self_init128_kernel<<<gNF, TB, 0, stream>>>(bufA, dinv, bufB, N);
    edge_agg128_kernel<<<E, 128, 0, stream>>>(bufA, src, dst, dinv, bufB, E);
    bn_relu_kernel<<<gNF, TB, 0, stream>>>(bufB, b1, g1, be1, m1, v1, bufA, N);

    // Layer 2
    gemm128_wmma_kernel<<<gTiles, 256, 0, stream>>>(bufA, W2, bufB, N);
    self_init128_kernel<<<gNF, TB, 0, stream>>>(bufB, dinv, bufA, N);
    edge_agg128_kernel<<<E, 128, 0, stream>>>(bufB, src, dst, dinv, bufA, E);
    bn_relu_kernel<<<gNF, TB, 0, stream>>>(bufA, b2, g2, be2, m2, v2, bufB, N);

    // Layer 3 + log_softmax (aggregate straight into d_out)
    gemm_w3_kernel<<<gWave8, 256, 0, stream>>>(bufB, W3, h3, N);
    self_init2_kernel<<<gN, TB, 0, stream>>>(h3, dinv, outp, N);
    edge_agg2_kernel<<<gE, TB, 0, stream>>>(h3, src, dst, dinv, outp, E);
    logsoftmax2_kernel<<<gN, TB, 0, stream>>>(outp, b3, N);
}